// Mixer_40200893890844
// MI455X (gfx1250) — compile-verified
//
#include <hip/hip_runtime.h>
#include <hip/hip_bf16.h>

#define D 1024
#define H 4096
#define EPS 1e-5f

typedef __attribute__((ext_vector_type(2))) float v2f;
typedef __attribute__((ext_vector_type(8))) float v8f;

// epilogue ops
#define OP_NONE     0
#define OP_SIGMOID  1
#define OP_SQRELU   2
#define OP_ADDRES   3
#define OP_GATEDRES 4

// ---------------------------------------------------------------------------
// LayerNorm + time-mix. One block of 256 threads handles the full D=1024 vec.
// xv_out / tmv may be null (FFN path mixes only k and r).
// ---------------------------------------------------------------------------
__global__ __launch_bounds__(256) void ln_mix_kernel(
    const float* __restrict__ xin, const float* __restrict__ w,
    const float* __restrict__ b,   const float* __restrict__ prev,
    const float* __restrict__ tmk, const float* __restrict__ tmv,
    const float* __restrict__ tmr,
    float* __restrict__ xln_out, float* __restrict__ xk_out,
    float* __restrict__ xv_out,  float* __restrict__ xr_out)
{
    __shared__ float ssum[8];
    __shared__ float ssq[8];
    const int t = threadIdx.x;
    const int wave = t >> 5;
    const int lane = t & 31;

    float s = 0.f, sq = 0.f;
    for (int i = t; i < D; i += 256) {
        float v = xin[i];
        s += v;
        sq += v * v;
    }
    // deterministic wave32 shuffle reduction
    for (int off = 16; off > 0; off >>= 1) {
        s  += __shfl_xor(s,  off, 32);
        sq += __shfl_xor(sq, off, 32);
    }
    if (lane == 0) { ssum[wave] = s; ssq[wave] = sq; }
    __syncthreads();
    if (wave == 0) {
        float a = (lane < 8) ? ssum[lane] : 0.f;
        float c = (lane < 8) ? ssq[lane]  : 0.f;
        for (int off = 4; off > 0; off >>= 1) {
            a += __shfl_xor(a, off, 32);
            c += __shfl_xor(c, off, 32);
        }
        if (lane == 0) { ssum[0] = a; ssq[0] = c; }
    }
    __syncthreads();

    const float mu   = ssum[0] * (1.f / D);
    const float var  = ssq[0] * (1.f / D) - mu * mu;
    const float rstd = rsqrtf(var + EPS);

    for (int i = t; i < D; i += 256) {
        float xl = (xin[i] - mu) * rstd * w[i] + b[i];
        xln_out[i] = xl;
        float p = prev[i];
        float k = tmk[i];
        xk_out[i] = xl * k + p * (1.f - k);
        if (xv_out) {
            float v = tmv[i];
            xv_out[i] = xl * v + p * (1.f - v);
        }
        float r = tmr[i];
        xr_out[i] = xl * r + p * (1.f - r);
    }
}

// ---------------------------------------------------------------------------
// Multi-job WMMA mat-vec. Up to 3 independent jobs share one grid so their
// weight streams overlap on HBM (same-stream kernels serialize otherwise).
//
// y[0:rows] = op(W[rows x K] @ x) via V_WMMA_F32_16X16X4_F32 (fp32-exact).
// One block = 256 threads = 8 waves owning one 16-row tile; each wave
// accumulates a K/8 slice with 4 independent WMMA accumulators (breaks the
// WMMA->WMMA RAW chain); partials reduce through LDS.
//
// fp32 WMMA layout (ISA 7.12.2):
//   A (16x4): lanes 0-15 -> row m, K=0..1 in v0..v1 ; lanes 16-31 -> K=2..3
//   B (4x16): lanes 0-15 -> col n, K=0..1 in v0..v1 ; lanes 16-31 -> K=2..3
//   D (16x16): VGPR j holds row j (lanes 0-15) / row j+8 (lanes 16-31)
// The vector is broadcast into ALL 16 B columns (each lane loads the same
// x[k+2*half] pair) so the loop has no EXEC masking and every D column holds
// the mat-vec result; lane 0/16 extract rows 0-7 / 8-15.
// ---------------------------------------------------------------------------
struct MatvecJob {
    const float* W;
    const float* x;
    float*       y;
    const float* res;
    const float* gate;
    int K;        // reduction length (= ldw here)
    int ldw;
    int op;
    int nblocks;  // rows / 16
};

__device__ __forceinline__ void matvec_body(
    const float* __restrict__ W, const float* __restrict__ x,
    float* __restrict__ y, int K, int ldw, int op,
    const float* __restrict__ res, const float* __restrict__ gate, int blk)
{
    __shared__ float part[8][16];

    const int wave = threadIdx.x >> 5;
    const int lane = threadIdx.x & 31;
    const int half = lane >> 4;          // 0: K 0..1 ; 1: K 2..3
    const int m    = lane & 15;          // row within tile
    const int row0 = blk << 4;

    const int kchunk = K >> 3;           // K/8 per wave (multiple of 16 here)
    const int k0 = wave * kchunk;
    const int k1 = k0 + kchunk;

    const float* wrow = W + (size_t)(row0 + m) * (size_t)ldw + (half << 1);
    const float* xrow = x + (half << 1);

    v8f c0 = {}; v8f c1 = {}; v8f c2 = {}; v8f c3 = {};

    for (int k = k0; k < k1; k += 16) {
        v2f a0 = *(const v2f*)(wrow + k);
        v2f a1 = *(const v2f*)(wrow + k + 4);
        v2f a2 = *(const v2f*)(wrow + k + 8);
        v2f a3 = *(const v2f*)(wrow + k + 12);
        // broadcast vector into every B column (no divergence, no zero-fill)
        v2f b0 = *(const v2f*)(xrow + k);
        v2f b1 = *(const v2f*)(xrow + k + 4);
        v2f b2 = *(const v2f*)(xrow + k + 8);
        v2f b3 = *(const v2f*)(xrow + k + 12);
        // (neg_a, A, neg_b, B, c_mod, C, reuse_a, reuse_b)
        c0 = __builtin_amdgcn_wmma_f32_16x16x4_f32(false, a0, false, b0, (short)0, c0, false, false);
        c1 = __builtin_amdgcn_wmma_f32_16x16x4_f32(false, a1, false, b1, (short)0, c1, false, false);
        c2 = __builtin_amdgcn_wmma_f32_16x16x4_f32(false, a2, false, b2, (short)0, c2, false, false);
        c3 = __builtin_amdgcn_wmma_f32_16x16x4_f32(false, a3, false, b3, (short)0, c3, false, false);
    }

    v8f c = c0 + c1 + c2 + c3;

    // lane 0 holds rows 0..7 in c[0..7], lane 16 rows 8..15 (column 0)
    if (m == 0) {
        const int rbase = half << 3;
        #pragma unroll
        for (int j = 0; j < 8; ++j)
            part[wave][rbase + j] = c[j];
    }
    __syncthreads();

    if (threadIdx.x < 16) {
        float v = 0.f;
        #pragma unroll
        for (int wv = 0; wv < 8; ++wv)
            v += part[wv][threadIdx.x];
        const int r = row0 + threadIdx.x;
        switch (op) {
            case OP_SIGMOID:  v = 1.f / (1.f + expf(-v));              break;
            case OP_SQRELU:   { float t = fmaxf(v, 0.f); v = t * t; }  break;
            case OP_ADDRES:   v = res[r] + v;                          break;
            case OP_GATEDRES: v = res[r] + gate[r] * v;                break;
            default: break;
        }
        y[r] = v;
    }
}

__global__ __launch_bounds__(256) void wmma_matvec3_kernel(
    MatvecJob j0, MatvecJob j1, MatvecJob j2)
{
    int blk = blockIdx.x;
    if (blk < j0.nblocks) {
        matvec_body(j0.W, j0.x, j0.y, j0.K, j0.ldw, j0.op, j0.res, j0.gate, blk);
        return;
    }
    blk -= j0.nblocks;
    if (blk < j1.nblocks) {
        matvec_body(j1.W, j1.x, j1.y, j1.K, j1.ldw, j1.op, j1.res, j1.gate, blk);
        return;
    }
    blk -= j1.nblocks;
    matvec_body(j2.W, j2.x, j2.y, j2.K, j2.ldw, j2.op, j2.res, j2.gate, blk);
}

// ---------------------------------------------------------------------------
// Elementwise WKV single-step. Consumes k, v, r(sigmoided), state rows 2..4.
// Emits rwkv = r*wkv (for ow matvec) and new states s2,s3,s4.
// ---------------------------------------------------------------------------
__global__ __launch_bounds__(256) void wkv_kernel(
    const float* __restrict__ kvec, const float* __restrict__ vvec,
    const float* __restrict__ rvec, const float* __restrict__ state,
    const float* __restrict__ time_first, const float* __restrict__ time_decay,
    float* __restrict__ rwkv,
    float* __restrict__ s2, float* __restrict__ s3, float* __restrict__ s4)
{
    const int i = blockIdx.x * 256 + threadIdx.x;
    if (i >= D) return;

    float k  = kvec[i];
    float v  = vvec[i];
    float r  = rvec[i];
    float aa = state[2 * D + i];
    float bb = state[3 * D + i];
    float pp = state[4 * D + i];

    float ww = time_first[i] + k;
    float qq = fmaxf(pp, ww);
    float e1 = expf(pp - qq);
    float e2 = expf(ww - qq);
    float wkv = (e1 * aa + e2 * v) / (e1 * bb + e2);
    rwkv[i] = r * wkv;

    float ww2 = pp + time_decay[i];
    float qq2 = fmaxf(ww2, k);
    float f1 = expf(ww2 - qq2);
    float f2 = expf(k - qq2);
    s2[i] = f1 * aa + f2 * v;
    s3[i] = f1 * bb + f2;
    s4[i] = qq2;
}

// ---------------------------------------------------------------------------
extern "C" void kernel_launch(void* const* d_in, const int* in_sizes, int n_in,
                              void* d_out, int out_size, void* d_ws, size_t ws_size,
                              hipStream_t stream) {
    const float* x     = (const float*)d_in[0];
    const float* state = (const float*)d_in[1];
    const float* tmk   = (const float*)d_in[2];
    const float* tmv   = (const float*)d_in[3];
    const float* tmr   = (const float*)d_in[4];
    const float* tf    = (const float*)d_in[5];
    const float* td    = (const float*)d_in[6];
    const float* kw    = (const float*)d_in[7];
    const float* vw    = (const float*)d_in[8];
    const float* rw    = (const float*)d_in[9];
    const float* ow    = (const float*)d_in[10];
    const float* lnw   = (const float*)d_in[11];
    const float* lnb   = (const float*)d_in[12];
    const float* ftmk  = (const float*)d_in[13];
    const float* ftmr  = (const float*)d_in[14];
    const float* fkw   = (const float*)d_in[15];
    const float* fvw   = (const float*)d_in[16];
    const float* frw   = (const float*)d_in[17];
    const float* flnw  = (const float*)d_in[18];
    const float* flnb  = (const float*)d_in[19];

    float* ws    = (float*)d_ws;
    float* xk    = ws + 1 * D;
    float* xv    = ws + 2 * D;
    float* xr    = ws + 3 * D;
    float* kvec  = ws + 4 * D;
    float* vvec  = ws + 5 * D;
    float* rvec  = ws + 6 * D;
    float* rwkv  = ws + 7 * D;
    float* x_att = ws + 8 * D;
    float* fxk   = ws + 9 * D;
    float* fxr   = ws + 10 * D;
    float* fr    = ws + 11 * D;
    float* fk    = ws + 12 * D;   // H = 4096 floats

    float* out = (float*)d_out;   // out[0:D], then s0..s4
    float* s0  = out + 1 * D;
    float* s1  = out + 2 * D;
    float* s2  = out + 3 * D;
    float* s3  = out + 4 * D;
    float* s4  = out + 5 * D;

    const MatvecJob jnull = { nullptr, nullptr, nullptr, nullptr, nullptr, 0, 0, OP_NONE, 0 };

    // 1. LN(x) + time-mix -> xln(=s1), xk, xv, xr
    ln_mix_kernel<<<1, 256, 0, stream>>>(x, lnw, lnb, state + 1 * D,
                                         tmk, tmv, tmr, s1, xk, xv, xr);

    // 2. r/k/v mat-vecs in ONE launch (independent weight streams overlap)
    {
        MatvecJob jk = { kw, xk, kvec, nullptr, nullptr, D, D, OP_NONE,    D / 16 };
        MatvecJob jv = { vw, xv, vvec, nullptr, nullptr, D, D, OP_NONE,    D / 16 };
        MatvecJob jr = { rw, xr, rvec, nullptr, nullptr, D, D, OP_SIGMOID, D / 16 };
        wmma_matvec3_kernel<<<3 * (D / 16), 256, 0, stream>>>(jk, jv, jr);
    }

    // 3. WKV elementwise -> rwkv, s2, s3, s4
    wkv_kernel<<<D / 256, 256, 0, stream>>>(kvec, vvec, rvec, state, tf, td,
                                            rwkv, s2, s3, s4);

    // 4. x_att = x + ow @ rwkv
    {
        MatvecJob jo = { ow, rwkv, x_att, x, nullptr, D, D, OP_ADDRES, D / 16 };
        wmma_matvec3_kernel<<<D / 16, 256, 0, stream>>>(jo, jnull, jnull);
    }

    // 5. LN(x_att) + FFN time-mix -> x2(=s0), fxk, fxr
    ln_mix_kernel<<<1, 256, 0, stream>>>(x_att, flnw, flnb, state /* state[0] */,
                                         ftmk, nullptr, ftmr, s0, fxk, nullptr, fxr);

    // 6. fr = sigmoid(ffn_rw@fxr) and fk = relu(ffn_kw@fxk)^2 in ONE launch
    {
        MatvecJob jfr = { frw, fxr, fr, nullptr, nullptr, D, D, OP_SIGMOID, D / 16 };
        MatvecJob jfk = { fkw, fxk, fk, nullptr, nullptr, D, D, OP_SQRELU,  H / 16 };
        wmma_matvec3_kernel<<<(D / 16) + (H / 16), 256, 0, stream>>>(jfr, jfk, jnull);
    }

    // 7. out = x_att + fr * (ffn_vw @ fk)
    {
        MatvecJob jout = { fvw, fk, out, x_att, fr, H, H, OP_GATEDRES, D / 16 };
        wmma_matvec3_kernel<<<D / 16, 256, 0, stream>>>(jout, jnull, jnull);
    }
}